// ContextualLSTMCell_74955769250717
// MI455X (gfx1250) — compile-verified
//
#include <hip/hip_runtime.h>
#include <hip/hip_bf16.h>
#include <math.h>

typedef __attribute__((ext_vector_type(2))) float v2f;
typedef __attribute__((ext_vector_type(8))) float v8f;

#define BSZ 16
#define ESZ 1024
#define HSZ 1024
#define H4  4096

// --- fast gate math on CDNA5 trans units -----------------------------------
__device__ __forceinline__ float fast_tanh(float x) {
    float r;
    // V_TANH_F32 is a gfx1250 transcendental; v_nop covers the 1-op TRANS
    // result hazard (ISA 7.4) without relying on compiler knowledge of asm.
    asm("v_tanh_f32 %0, %1\n\tv_nop" : "=v"(r) : "v"(x));
    return r;
}
__device__ __forceinline__ float fast_sigmoid(float x) {
    // 1/(1+e^-x) = rcp(1 + exp2(-x*log2e))  -> v_exp_f32 + v_rcp_f32
    float t = __builtin_amdgcn_exp2f(x * -1.44269504088896340736f);
    return __builtin_amdgcn_rcpf(1.0f + t);
}

// ---------------------------------------------------------------------------
// Kernel 1: ifgo[16,4096] = [x|h](16x2048) @ [Wi;Wh](2048x4096) + Wi_b + Wh_b
// grid = 4096/16 = 256 column tiles; block = 512 threads = 16 wave32 waves.
// Wave w owns combined-K chunk [w*128, w*128+128): w<8 -> x@Wi, w>=8 -> h@Wh.
// 4096 waves total: enough loads in flight to feed 23.3 TB/s HBM.
// ---------------------------------------------------------------------------
__global__ __launch_bounds__(512)
void ifgo_wmma_kernel(const float* __restrict__ x,
                      const float* __restrict__ h,
                      const float* __restrict__ Wi,
                      const float* __restrict__ Wi_b,
                      const float* __restrict__ Wh,
                      const float* __restrict__ Wh_b,
                      float* __restrict__ ifgo)
{
    __shared__ float lds[16][256];   // 16 waves x (16x16 f32 partial tile)

    const int tid  = threadIdx.x;
    const int lane = tid & 31;
    const int w    = tid >> 5;           // 0..15 : K chunk
    const int n0   = blockIdx.x * 16;    // column tile in 4H

    const int m  = lane & 15;            // A row (batch) / B col-in-tile
    const int kp = (lane >> 4) << 1;     // K sub-pair: 0 or 2

    const float* Ap = (w < 8) ? x  : h;  // E == H == 1024: same row stride
    const float* Bp = (w < 8) ? Wi : Wh;
    const int kbase = (w & 7) * 128;
    const int bcol  = n0 + m;

    const float* arow = Ap + m * ESZ;

    v8f acc = {};
    #pragma unroll 8
    for (int k = kbase; k < kbase + 128; k += 4) {
        const int ka = k + kp;
        // A fragment: lane holds A[m][ka], A[m][ka+1]  (global_load_b64, L2-hot)
        v2f a = *(const v2f*)(arow + ka);
        // B fragment: lane holds W[ka][bcol], W[ka+1][bcol] (streams HBM once)
        v2f b;
        b.x = Bp[(size_t)ka * H4 + bcol];
        b.y = Bp[(size_t)(ka + 1) * H4 + bcol];
        acc = __builtin_amdgcn_wmma_f32_16x16x4_f32(false, a, false, b,
                                                    (short)0, acc, false, false);
    }

    // spill partial tile: C/D layout -> element (r + (lane/16)*8, lane%16)
    {
        float* t = &lds[w][0];
        const int mh = (lane >> 4) << 3;
        #pragma unroll
        for (int r = 0; r < 8; ++r)
            t[(r + mh) * 16 + m] = acc[r];
    }
    __syncthreads();

    // Deterministic 16-way reduction (fixed order) + bias -> ifgo tile.
    if (tid < 256) {
        const int bm  = tid >> 4;        // batch row 0..15
        const int n   = tid & 15;
        const int col = n0 + n;          // column in 4H
        float s = Wi_b[col] + Wh_b[col];
        #pragma unroll
        for (int cc = 0; cc < 16; ++cc)
            s += lds[cc][bm * 16 + n];
        ifgo[bm * H4 + col] = s;
    }
}

// ---------------------------------------------------------------------------
// Kernel 2: elementwise LSTM gating. 16x1024 outputs; ifgo is L2-resident.
// ---------------------------------------------------------------------------
__global__ __launch_bounds__(256)
void lstm_gate_kernel(const float* __restrict__ ifgo,
                      const float* __restrict__ c,
                      float* __restrict__ out)
{
    const int idx = blockIdx.x * 256 + threadIdx.x;   // 0 .. 16*1024-1
    const int bm  = idx >> 10;
    const int col = idx & 1023;
    const int base = bm * H4 + col;

    const float ig = fast_sigmoid(ifgo[base]);
    const float fg = fast_sigmoid(ifgo[base + HSZ]);
    const float gg = fast_tanh   (ifgo[base + 2 * HSZ]);
    const float og = fast_sigmoid(ifgo[base + 3 * HSZ]);

    const float cn = fg * c[bm * HSZ + col] + ig * gg;
    const float hn = og * fast_tanh(cn);

    out[bm * HSZ + col]             = hn;   // h_new (tuple slot 0)
    out[BSZ * HSZ + bm * HSZ + col] = cn;   // c_new (tuple slot 1)
}

extern "C" void kernel_launch(void* const* d_in, const int* in_sizes, int n_in,
                              void* d_out, int out_size, void* d_ws, size_t ws_size,
                              hipStream_t stream) {
    // setup_inputs order: x, h, c, context, Wi, Wi_b, Wh, Wh_b, AZ_il, AZ_ir, AZ_hl, AZ_hr
    const float* x    = (const float*)d_in[0];
    const float* h    = (const float*)d_in[1];
    const float* c    = (const float*)d_in[2];
    // d_in[3] (context) and d_in[8..11] (AZ_*) are dead: reference scales them by 0.
    const float* Wi   = (const float*)d_in[4];
    const float* Wi_b = (const float*)d_in[5];
    const float* Wh   = (const float*)d_in[6];
    const float* Wh_b = (const float*)d_in[7];
    float* out  = (float*)d_out;
    float* ifgo = (float*)d_ws;            // 16*4096 f32 = 256 KB scratch

    ifgo_wmma_kernel<<<dim3(H4 / 16), dim3(512), 0, stream>>>(
        x, h, Wi, Wi_b, Wh, Wh_b, ifgo);
    lstm_gate_kernel<<<dim3((BSZ * HSZ) / 256), dim3(256), 0, stream>>>(
        ifgo, c, out);
}